// RiemannianBatchNorm_60318520705590
// MI455X (gfx1250) — compile-verified
//
#include <hip/hip_runtime.h>

typedef float v2f __attribute__((ext_vector_type(2)));
typedef float v4f __attribute__((ext_vector_type(4)));
typedef float v8f __attribute__((ext_vector_type(8)));

// explicit address spaces: AS3 -> ds_* ops, AS1 -> global_* ops
typedef __attribute__((address_space(3))) float lfloat;
typedef __attribute__((address_space(1))) float gfloat;
typedef __attribute__((address_space(1))) v4f   gv4f;

namespace rbn {

__device__ __forceinline__ lfloat* lptr(float* p) {
  // generic LDS address: low 32 bits are the LDS byte offset (ISA 10.2)
  return (lfloat*)(unsigned int)(unsigned long long)p;
}
__device__ __forceinline__ gfloat* gptr(float* p) {
  return (gfloat*)(unsigned long long)p;
}
__device__ __forceinline__ const gfloat* gptr(const float* p) {
  return (const gfloat*)(unsigned long long)p;
}

constexpr int BATCH = 32768;
constexpr int MAT   = 1024;       // 32*32
constexpr int LDAP  = 34;         // row stride: 8B-aligned rows, conflict-free
constexpr int MBUF  = 32 * LDAP;  // floats per matrix buffer

constexpr int NS_ITERS   = 5;     // Newton-Schulz iterations per sqrt
constexpr int LOG_LEVELS = 4;     // inverse scaling-and-squaring depth (2^4)
constexpr int EXP_SQUARES = 5;    // expm scaling-and-squaring depth

constexpr int ABLK  = 64;                 // mean-partial blocks
constexpr int CBLK  = 256;                // heavy kernels: blocks of 2 waves
constexpr int WAVES = CBLK * 2;           // 512 worker waves
constexpr int PER   = BATCH / WAVES;      // 64 matrices per wave

// workspace layout (float offsets)
constexpr size_t WS_MEAN0  = 0;
constexpr size_t WS_MS0    = WS_MEAN0 + MAT;
constexpr size_t WS_MIS0   = WS_MS0   + MAT;
constexpr size_t WS_TMEAN  = WS_MIS0  + MAT;
constexpr size_t WS_MIS1   = WS_TMEAN + MAT;
constexpr size_t WS_P      = WS_MIS1  + MAT;             // 32 floats, [0] = p
constexpr size_t WS_PARTA  = WS_P     + 32;              // ABLK * MAT
constexpr size_t WS_PARTC  = WS_PARTA + (size_t)ABLK * MAT;   // WAVES * MAT
constexpr size_t WS_PARTD2 = WS_PARTC + (size_t)WAVES * MAT;  // WAVES floats

// ---------------------------------------------------------------------------
// 32x32 fp32 matmul on one wave via V_WMMA_F32_16X16X4_F32, operands in LDS.
// D may alias A and/or B: all source fragments are consumed into register
// accumulators before any store issues; per-wave DS ordering does the rest.
// ---------------------------------------------------------------------------
__device__ __attribute__((always_inline)) inline void mm32(lfloat* D, const lfloat* A,
                                                           const lfloat* B, int lane) {
  const int half = (lane >> 4) & 1;   // K-half selector per ISA A/B layout
  const int l16  = lane & 15;
  v8f acc00 = {}, acc01 = {}, acc10 = {}, acc11 = {};
#pragma unroll
  for (int kc = 0; kc < 32; kc += 4) {
    v2f a0, a1, b0, b1;
    // A 16x4 fragment: VGPR0/1 hold K = 2*half, 2*half+1 for row M=l16 (8B aligned pair)
    a0.x = A[l16 * LDAP + kc + 2 * half];
    a0.y = A[l16 * LDAP + kc + 2 * half + 1];
    a1.x = A[(l16 + 16) * LDAP + kc + 2 * half];
    a1.y = A[(l16 + 16) * LDAP + kc + 2 * half + 1];
    // B 4x16 fragment: rows K = 2*half, 2*half+1, col N=l16
    b0.x = B[(kc + 2 * half) * LDAP + l16];
    b0.y = B[(kc + 2 * half + 1) * LDAP + l16];
    b1.x = B[(kc + 2 * half) * LDAP + l16 + 16];
    b1.y = B[(kc + 2 * half + 1) * LDAP + l16 + 16];
    acc00 = __builtin_amdgcn_wmma_f32_16x16x4_f32(false, a0, false, b0, (short)0, acc00, false, false);
    acc01 = __builtin_amdgcn_wmma_f32_16x16x4_f32(false, a0, false, b1, (short)0, acc01, false, false);
    acc10 = __builtin_amdgcn_wmma_f32_16x16x4_f32(false, a1, false, b0, (short)0, acc10, false, false);
    acc11 = __builtin_amdgcn_wmma_f32_16x16x4_f32(false, a1, false, b1, (short)0, acc11, false, false);
  }
  // C/D layout: VGPR r -> row r + 8*half, col l16 (per 16x16 tile)
#pragma unroll
  for (int r = 0; r < 8; ++r) {
    D[(r + 8 * half) * LDAP + l16]            = acc00[r];
    D[(r + 8 * half) * LDAP + l16 + 16]       = acc01[r];
    D[(r + 8 * half + 16) * LDAP + l16]       = acc10[r];
    D[(r + 8 * half + 16) * LDAP + l16 + 16]  = acc11[r];
  }
}

// max row-sum norm (Gershgorin bound on lambda_max); all lanes return same value
__device__ inline float rownorm(const lfloat* A, lfloat* red, int lane) {
  float s = 0.0f;
  for (int c = 0; c < 32; ++c) s += fabsf(A[lane * LDAP + c]);
  red[lane] = s;
  float m = 0.0f;
  for (int i = 0; i < 32; ++i) m = fmaxf(m, (float)red[i]);
  return m;
}

// Y = sqrtm(S), Z = isqrtm(S) via scaled coupled Newton-Schulz (spectrum/c in (0,1])
__device__ inline void mat_sqrt(const lfloat* S, lfloat* Y, lfloat* Z, lfloat* T,
                                lfloat* red, int lane) {
  float c  = rownorm(S, red, lane);
  float rc = 1.0f / c;
  for (int col = 0; col < 32; ++col) {
    Y[lane * LDAP + col] = S[lane * LDAP + col] * rc;
    Z[lane * LDAP + col] = (col == lane) ? 1.0f : 0.0f;
  }
#pragma unroll 1
  for (int it = 0; it < NS_ITERS; ++it) {
    mm32(T, Z, Y, lane);                       // T = Z*Y
    for (int col = 0; col < 32; ++col)
      T[lane * LDAP + col] = ((col == lane) ? 1.5f : 0.0f) - 0.5f * T[lane * LDAP + col];
    mm32(Y, Y, T, lane);                       // Y <- Y*T
    mm32(Z, T, Z, lane);                       // Z <- T*Z
  }
  float sc = sqrtf(c), rsc = 1.0f / sc;
  for (int col = 0; col < 32; ++col) {
    Y[lane * LDAP + col] *= sc;
    Z[lane * LDAP + col] *= rsc;
  }
}

// L = logm(S) by inverse scaling-and-squaring; destroys S, uses Y,Z,T
__device__ inline void mat_logm(lfloat* S, lfloat* L, lfloat* Y, lfloat* Z,
                                lfloat* T, lfloat* red, int lane) {
#pragma unroll 1
  for (int j = 0; j < LOG_LEVELS; ++j) {
    mat_sqrt(S, Y, Z, T, red, lane);
    for (int col = 0; col < 32; ++col) S[lane * LDAP + col] = Y[lane * LDAP + col];
  }
  // E = S - I (in Y); log(I+E) ~= E(c1 I + E(c2 I + ... + E(c5 I + c6 E)))
  for (int col = 0; col < 32; ++col)
    Y[lane * LDAP + col] = S[lane * LDAP + col] - ((col == lane) ? 1.0f : 0.0f);
  for (int col = 0; col < 32; ++col)
    T[lane * LDAP + col] = (-1.0f / 6.0f) * Y[lane * LDAP + col] + ((col == lane) ? 0.2f : 0.0f);
  const float ck[4] = {-0.25f, 1.0f / 3.0f, -0.5f, 1.0f};
#pragma unroll 1
  for (int kk = 0; kk < 4; ++kk) {
    mm32(T, Y, T, lane);
    T[lane * LDAP + lane] += ck[kk];
  }
  mm32(L, Y, T, lane);
  const float two_k = (float)(1 << LOG_LEVELS);
  for (int col = 0; col < 32; ++col) L[lane * LDAP + col] *= two_k;
}

// T = expm(U) via scaling-and-squaring + degree-8 Horner Taylor; destroys U
__device__ inline void mat_expm(lfloat* U, lfloat* T, int lane) {
  const float sc = 1.0f / (float)(1 << EXP_SQUARES);
  for (int col = 0; col < 32; ++col) U[lane * LDAP + col] *= sc;
  for (int col = 0; col < 32; ++col)
    T[lane * LDAP + col] = 0.125f * U[lane * LDAP + col] + ((col == lane) ? 1.0f : 0.0f);
#pragma unroll 1
  for (int k = 7; k >= 1; --k) {
    mm32(T, U, T, lane);
    float rk = 1.0f / (float)k;
    for (int col = 0; col < 32; ++col)
      T[lane * LDAP + col] = T[lane * LDAP + col] * rk + ((col == lane) ? 1.0f : 0.0f);
  }
#pragma unroll 1
  for (int j = 0; j < EXP_SQUARES; ++j) mm32(T, T, T, lane);
}

// ------------------------------- kernels -----------------------------------

__global__ __launch_bounds__(256) void kmean_part(const float* __restrict__ x,
                                                  float* __restrict__ part) {
  const gfloat* xg = gptr(x);
  gfloat* pg = gptr(part);
  const int b = blockIdx.x, t = threadIdx.x;
  const int SLICE = BATCH / ABLK;
  float acc[4] = {0.f, 0.f, 0.f, 0.f};
  for (int i = 0; i < SLICE; ++i) {
    const gfloat* m = xg + (size_t)(b * SLICE + i) * MAT;
#pragma unroll
    for (int j = 0; j < 4; ++j) acc[j] += m[t + 256 * j];
  }
#pragma unroll
  for (int j = 0; j < 4; ++j) pg[(size_t)b * MAT + t + 256 * j] = acc[j];
}

__global__ __launch_bounds__(256) void kmean_reduce(const float* __restrict__ part,
                                                    float* __restrict__ mean0) {
  const gfloat* pg = gptr(part);
  const int e = blockIdx.x * 256 + threadIdx.x;  // 1024 threads
  float s = 0.f;
  for (int b = 0; b < ABLK; ++b) s += pg[(size_t)b * MAT + e];
  gptr(mean0)[e] = s * (1.0f / (float)BATCH);
}

__global__ __launch_bounds__(32) void ksqrt_mean(const float* __restrict__ mean0,
                                                 float* __restrict__ ms0,
                                                 float* __restrict__ mis0) {
  __shared__ float Sb[MBUF], Yb[MBUF], Zb[MBUF], Tb[MBUF], redb[32];
  lfloat *S = lptr(Sb), *Y = lptr(Yb), *Z = lptr(Zb), *T = lptr(Tb), *red = lptr(redb);
  const int lane = threadIdx.x;
  const gfloat* m0 = gptr(mean0);
  for (int col = 0; col < 32; ++col) S[lane * LDAP + col] = m0[lane * 32 + col];
  mat_sqrt(S, Y, Z, T, red, lane);
  gfloat* so = gptr(ms0);
  gfloat* io = gptr(mis0);
  for (int col = 0; col < 32; ++col) {
    so[lane * 32 + col] = Y[lane * LDAP + col];
    io[lane * 32 + col] = Z[lane * LDAP + col];
  }
}

__global__ __launch_bounds__(64) void klog_accum(const float* __restrict__ x,
                                                 const float* __restrict__ mis0,
                                                 float* __restrict__ partC) {
  __shared__ float misb[MBUF];
  __shared__ float bufsb[2][6][MBUF];  // per wave: S, Y, Z, T, L, ACC
  __shared__ float redb[2][32];
  const int t = threadIdx.x, lane = t & 31, w = t >> 5;
  lfloat* mis = lptr(misb);
  const gfloat* misg = gptr(mis0);
  for (int i = t; i < MAT; i += 64) mis[(i >> 5) * LDAP + (i & 31)] = misg[i];
  __syncthreads();
  lfloat* base = lptr(&bufsb[0][0][0]) + (size_t)w * 6 * MBUF;
  lfloat *S = base, *Y = base + MBUF, *Z = base + 2 * MBUF;
  lfloat *T = base + 3 * MBUF, *L = base + 4 * MBUF, *ACC = base + 5 * MBUF;
  lfloat* red = lptr(&redb[0][0]) + w * 32;
  for (int col = 0; col < 32; ++col) ACC[lane * LDAP + col] = 0.0f;
  const int wave = blockIdx.x * 2 + w;
  const gfloat* xg = gptr(x);
#pragma unroll 1
  for (int k = 0; k < PER; ++k) {
    const int i = wave + k * WAVES;
    const gv4f* xr = (const gv4f*)(xg + (size_t)i * MAT + lane * 32);
#pragma unroll
    for (int q = 0; q < 8; ++q) {
      v4f v = xr[q];
      S[lane * LDAP + 4 * q + 0] = v.x; S[lane * LDAP + 4 * q + 1] = v.y;
      S[lane * LDAP + 4 * q + 2] = v.z; S[lane * LDAP + 4 * q + 3] = v.w;
    }
    mm32(Y, mis, S, lane);       // whiten: S <- mis * x * mis
    mm32(S, Y, mis, lane);
    mat_logm(S, L, Y, Z, T, red, lane);
    for (int col = 0; col < 32; ++col) ACC[lane * LDAP + col] += L[lane * LDAP + col];
  }
  gfloat* pc = gptr(partC) + (size_t)wave * MAT;
  for (int col = 0; col < 32; ++col) pc[lane * 32 + col] = ACC[lane * LDAP + col];
}

__global__ __launch_bounds__(256) void ktmean(const float* __restrict__ partC,
                                              float* __restrict__ tmean) {
  const gfloat* pg = gptr(partC);
  const int e = blockIdx.x * 256 + threadIdx.x;  // 1024 threads
  float s = 0.f;
  for (int wv = 0; wv < WAVES; ++wv) s += pg[(size_t)wv * MAT + e];
  gptr(tmean)[e] = s * (1.0f / (float)BATCH);
}

__global__ __launch_bounds__(32) void kupdate(const float* __restrict__ tmean,
                                              const float* __restrict__ ms0,
                                              float* __restrict__ mis1) {
  __shared__ float Ub[MBUF], Tb[MBUF], Ab[MBUF], Yb[MBUF], Zb[MBUF], Wb[MBUF], redb[32];
  lfloat *U = lptr(Ub), *T = lptr(Tb), *A = lptr(Ab), *Y = lptr(Yb);
  lfloat *Z = lptr(Zb), *W = lptr(Wb), *red = lptr(redb);
  const int lane = threadIdx.x;
  const gfloat* tg = gptr(tmean);
  for (int col = 0; col < 32; ++col) U[lane * LDAP + col] = tg[lane * 32 + col];
  mat_expm(U, T, lane);                       // T = expm(t_mean)
  const gfloat* sg = gptr(ms0);
  for (int col = 0; col < 32; ++col) A[lane * LDAP + col] = sg[lane * 32 + col];
  mm32(W, A, T, lane);                        // mean1 = ms0 * expm(t) * ms0
  mm32(U, W, A, lane);
  mat_sqrt(U, Y, Z, T, red, lane);            // Z = isqrtm(mean1)
  gfloat* io = gptr(mis1);
  for (int col = 0; col < 32; ++col) io[lane * 32 + col] = Z[lane * LDAP + col];
}

__global__ __launch_bounds__(64) void kwhiten_log(const float* __restrict__ x,
                                                  const float* __restrict__ mis1,
                                                  float* __restrict__ Lout,
                                                  float* __restrict__ partD2) {
  __shared__ float misb[MBUF];
  __shared__ float bufsb[2][5][MBUF];  // per wave: S, Y, Z, T, L
  __shared__ float redb[2][32];
  const int t = threadIdx.x, lane = t & 31, w = t >> 5;
  lfloat* mis = lptr(misb);
  const gfloat* misg = gptr(mis1);
  for (int i = t; i < MAT; i += 64) mis[(i >> 5) * LDAP + (i & 31)] = misg[i];
  __syncthreads();
  lfloat* base = lptr(&bufsb[0][0][0]) + (size_t)w * 5 * MBUF;
  lfloat *S = base, *Y = base + MBUF, *Z = base + 2 * MBUF;
  lfloat *T = base + 3 * MBUF, *L = base + 4 * MBUF;
  lfloat* red = lptr(&redb[0][0]) + w * 32;
  const int wave = blockIdx.x * 2 + w;
  const gfloat* xg = gptr(x);
  gfloat* lg = gptr(Lout);
  float d2 = 0.0f;
#pragma unroll 1
  for (int k = 0; k < PER; ++k) {
    const int i = wave + k * WAVES;
    const gv4f* xr = (const gv4f*)(xg + (size_t)i * MAT + lane * 32);
#pragma unroll
    for (int q = 0; q < 8; ++q) {
      v4f v = xr[q];
      S[lane * LDAP + 4 * q + 0] = v.x; S[lane * LDAP + 4 * q + 1] = v.y;
      S[lane * LDAP + 4 * q + 2] = v.z; S[lane * LDAP + 4 * q + 3] = v.w;
    }
    mm32(Y, mis, S, lane);
    mm32(S, Y, mis, lane);
    mat_logm(S, L, Y, Z, T, red, lane);
    gv4f* lr = (gv4f*)(lg + (size_t)i * MAT + lane * 32);
#pragma unroll
    for (int q = 0; q < 8; ++q) {
      v4f v;
      v.x = L[lane * LDAP + 4 * q + 0]; v.y = L[lane * LDAP + 4 * q + 1];
      v.z = L[lane * LDAP + 4 * q + 2]; v.w = L[lane * LDAP + 4 * q + 3];
      d2 += v.x * v.x + v.y * v.y + v.z * v.z + v.w * v.w;
      lr[q] = v;
    }
  }
  red[lane] = d2;
  float tot = 0.0f;
  for (int i = 0; i < 32; ++i) tot += (float)red[i];
  if (lane == 0) gptr(partD2)[wave] = tot;
}

__global__ __launch_bounds__(256) void kstd(const float* __restrict__ partD2,
                                            const float* __restrict__ scale,
                                            float* __restrict__ pout) {
  __shared__ float smb[256];
  lfloat* sm = lptr(smb);
  const gfloat* pg = gptr(partD2);
  const int t = threadIdx.x;
  sm[t] = pg[t] + pg[t + 256];
  __syncthreads();
  for (int s = 128; s > 0; s >>= 1) {
    if (t < s) sm[t] += sm[t + s];
    __syncthreads();
  }
  if (t == 0) {
    float stdv = sqrtf((float)sm[0] / (float)BATCH);
    gptr(pout)[0] = gptr(scale)[0] / (stdv + 1e-5f);
  }
}

__global__ __launch_bounds__(64) void kpow(float* __restrict__ Lio,
                                           const float* __restrict__ pscal) {
  __shared__ float bufsb[2][2][MBUF];  // per wave: U, T
  const int t = threadIdx.x, lane = t & 31, w = t >> 5;
  lfloat* base = lptr(&bufsb[0][0][0]) + (size_t)w * 2 * MBUF;
  lfloat *U = base, *T = base + MBUF;
  const float p = gptr(pscal)[0];
  const int wave = blockIdx.x * 2 + w;
  gfloat* lg = gptr(Lio);
#pragma unroll 1
  for (int k = 0; k < PER; ++k) {
    const int i = wave + k * WAVES;
    const gv4f* lr = (const gv4f*)(lg + (size_t)i * MAT + lane * 32);
#pragma unroll
    for (int q = 0; q < 8; ++q) {
      v4f v = lr[q];
      U[lane * LDAP + 4 * q + 0] = p * v.x; U[lane * LDAP + 4 * q + 1] = p * v.y;
      U[lane * LDAP + 4 * q + 2] = p * v.z; U[lane * LDAP + 4 * q + 3] = p * v.w;
    }
    mat_expm(U, T, lane);               // out = expm(p * logm(xt)) = xt^p
    gv4f* orow = (gv4f*)(lg + (size_t)i * MAT + lane * 32);
#pragma unroll
    for (int q = 0; q < 8; ++q) {
      v4f v;
      v.x = T[lane * LDAP + 4 * q + 0]; v.y = T[lane * LDAP + 4 * q + 1];
      v.z = T[lane * LDAP + 4 * q + 2]; v.w = T[lane * LDAP + 4 * q + 3];
      orow[q] = v;
    }
  }
}

}  // namespace rbn

extern "C" void kernel_launch(void* const* d_in, const int* in_sizes, int n_in,
                              void* d_out, int out_size, void* d_ws, size_t ws_size,
                              hipStream_t stream) {
  using namespace rbn;
  const float* x     = (const float*)d_in[0];
  const float* scale = (const float*)d_in[1];
  float* out = (float*)d_out;
  float* ws  = (float*)d_ws;
  (void)in_sizes; (void)n_in; (void)out_size; (void)ws_size;

  kmean_part <<<ABLK, 256, 0, stream>>>(x, ws + WS_PARTA);
  kmean_reduce<<<4,   256, 0, stream>>>(ws + WS_PARTA, ws + WS_MEAN0);
  ksqrt_mean <<<1,    32,  0, stream>>>(ws + WS_MEAN0, ws + WS_MS0, ws + WS_MIS0);
  klog_accum <<<CBLK, 64,  0, stream>>>(x, ws + WS_MIS0, ws + WS_PARTC);
  ktmean     <<<4,    256, 0, stream>>>(ws + WS_PARTC, ws + WS_TMEAN);
  kupdate    <<<1,    32,  0, stream>>>(ws + WS_TMEAN, ws + WS_MS0, ws + WS_MIS1);
  kwhiten_log<<<CBLK, 64,  0, stream>>>(x, ws + WS_MIS1, out, ws + WS_PARTD2);
  kstd       <<<1,    256, 0, stream>>>(ws + WS_PARTD2, scale, ws + WS_P);
  kpow       <<<CBLK, 64,  0, stream>>>(out, ws + WS_P);
}